// CapsuleLayer_70179765617004
// MI455X (gfx1250) — compile-verified
//
#include <hip/hip_runtime.h>

// Problem dims (CapsuleLayer reference): B=64, C=16, R=1152, I=64, O=64
#define RNODES 1152
#define BATCH  64
#define IDIM   64
#define ODIM   64
#define NCAPS  16
#define LSTRIDE 66   // padded LDS row stride (even for 8B align, stride%64==2 kills bank conflicts)

typedef __attribute__((ext_vector_type(2))) float v2f;
typedef __attribute__((ext_vector_type(8))) float v8f;

// ---------------------------------------------------------------------------
// Kernel 1: priors[c][b][r][o] = sum_i x[b][r][i] * W[c][r][i][o]
// One block per (c,r): a 64x64x64 fp32 GEMM done with v_wmma_f32_16x16x4_f32.
// 128 threads = 4 wave32; wave w computes output rows [16w,16w+16) x all 64 cols.
// ---------------------------------------------------------------------------
__global__ __launch_bounds__(128) void caps_priors_gemm(
    const float* __restrict__ x,       // (B, R, I)
    const float* __restrict__ w,       // (C, R, I, O)
    float* __restrict__ priors)        // (C, B, R, O)
{
    __shared__ float xa [BATCH * LSTRIDE];  // A tile: x[:, r, :]  as [b][i]
    __shared__ float wbT[ODIM  * LSTRIDE];  // B tile: W[c,r,:,:] transposed to [o][i]

    const int t = threadIdx.x;
    const int r = blockIdx.x % RNODES;
    const int c = blockIdx.x / RNODES;

    // Load A: 64 rows of 64 floats; rows strided by R*I in global. float4, coalesced.
    for (int idx = t; idx < BATCH * (IDIM / 4); idx += 128) {
        const int b   = idx >> 4;          // 16 float4 per row
        const int seg = idx & 15;
        const float4 v = ((const float4*)x)[((size_t)b * RNODES + r) * (IDIM / 4) + seg];
        float* dst = &xa[b * LSTRIDE + seg * 4];
        dst[0] = v.x; dst[1] = v.y; dst[2] = v.z; dst[3] = v.w;
    }
    // Load B: contiguous 4096 floats at W[c][r]; transpose [i][o] -> [o][i] into LDS.
    const size_t wbase = (size_t)(c * RNODES + r) * (IDIM * ODIM);
    for (int idx = t; idx < IDIM * (ODIM / 4); idx += 128) {
        const int i   = idx >> 4;          // source row (i), 16 float4 per row
        const int seg = idx & 15;
        const float4 v = ((const float4*)(w + wbase))[idx];
        wbT[(seg * 4 + 0) * LSTRIDE + i] = v.x;
        wbT[(seg * 4 + 1) * LSTRIDE + i] = v.y;
        wbT[(seg * 4 + 2) * LSTRIDE + i] = v.z;
        wbT[(seg * 4 + 3) * LSTRIDE + i] = v.w;
    }
    __syncthreads();

    const int wave = t >> 5;
    const int lane = t & 31;
    const int lm   = lane & 15;   // row/col within tile
    const int kh   = lane >> 4;   // k half: lanes 16-31 hold K=2,3 of each step
    const int m0   = wave * 16;   // output row base for this wave

    v8f acc0 = {}, acc1 = {}, acc2 = {}, acc3 = {};
    for (int kk = 0; kk < IDIM; kk += 4) {
        const int ko = kk + 2 * kh;
        // A frag: a[m][k]: lane = m + 16*(k>>1), vgpr = k&1  (ISA 7.12.2, 32-bit A 16x4)
        const v2f a  = *(const v2f*)&xa [(m0 + lm) * LSTRIDE + ko];
        // B frag (4x16, K x N), symmetric layout; wbT is o-major so it's contiguous
        const v2f b0 = *(const v2f*)&wbT[( 0 + lm) * LSTRIDE + ko];
        const v2f b1 = *(const v2f*)&wbT[(16 + lm) * LSTRIDE + ko];
        const v2f b2 = *(const v2f*)&wbT[(32 + lm) * LSTRIDE + ko];
        const v2f b3 = *(const v2f*)&wbT[(48 + lm) * LSTRIDE + ko];
        acc0 = __builtin_amdgcn_wmma_f32_16x16x4_f32(false, a, false, b0, (short)0, acc0, false, false);
        acc1 = __builtin_amdgcn_wmma_f32_16x16x4_f32(false, a, false, b1, (short)0, acc1, false, false);
        acc2 = __builtin_amdgcn_wmma_f32_16x16x4_f32(false, a, false, b2, (short)0, acc2, false, false);
        acc3 = __builtin_amdgcn_wmma_f32_16x16x4_f32(false, a, false, b3, (short)0, acc3, false, false);
    }

    // D layout: VGPR j, lanes 0-15 -> M=j, lanes 16-31 -> M=8+j; N = lane&15.
    #pragma unroll
    for (int j = 0; j < 8; j++) {
        const int row = m0 + j + 8 * kh;   // batch index
        float* dst = priors + ((size_t)(c * BATCH + row) * RNODES + r) * ODIM;
        dst[ 0 + lm] = acc0[j];
        dst[16 + lm] = acc1[j];
        dst[32 + lm] = acc2[j];
        dst[48 + lm] = acc3[j];
    }
}

// ---------------------------------------------------------------------------
// Kernel 2: dynamic routing (3 iterations) for one (c,b) pair per block.
// priors slice (1152x64 = 294KB) streamed from global; logits/probs in LDS.
// ---------------------------------------------------------------------------
__global__ __launch_bounds__(256) void caps_routing(
    const float* __restrict__ priors,       // (C, B, R, O)
    const float* __restrict__ route_logits, // (C, 1, R, 1, 1)
    float* __restrict__ out)                // (C, B, 1, 1, O)
{
    __shared__ float logit[RNODES];
    __shared__ float prob [RNODES];
    __shared__ float red  [256];
    __shared__ float spart[4 * ODIM];
    __shared__ float svec [ODIM];
    __shared__ float ovec [ODIM];

    const int t = threadIdx.x;
    const int b = blockIdx.x % BATCH;
    const int c = blockIdx.x / BATCH;
    const float* P = priors + (size_t)(c * BATCH + b) * RNODES * ODIM;

    for (int rr = t; rr < RNODES; rr += 256)
        logit[rr] = route_logits[c * RNODES + rr];
    __syncthreads();

    for (int iter = 0; iter < 3; iter++) {
        // ---- softmax over R ----
        float m = -3.402823466e38f;
        for (int rr = t; rr < RNODES; rr += 256) m = fmaxf(m, logit[rr]);
        red[t] = m; __syncthreads();
        for (int s = 128; s > 0; s >>= 1) {
            if (t < s) red[t] = fmaxf(red[t], red[t + s]);
            __syncthreads();
        }
        m = red[0]; __syncthreads();

        float psum = 0.f;
        for (int rr = t; rr < RNODES; rr += 256) {
            const float e = __expf(logit[rr] - m);
            prob[rr] = e;
            psum += e;
        }
        red[t] = psum; __syncthreads();
        for (int s = 128; s > 0; s >>= 1) {
            if (t < s) red[t] += red[t + s];
            __syncthreads();
        }
        const float inv = 1.f / red[0];
        __syncthreads();

        // ---- s[o] = inv * sum_r prob[r] * P[r][o]  (4 r-groups x 64 o-lanes) ----
        const int o = t & 63;
        const int g = t >> 6;
        float acc = 0.f;
        for (int rr = g; rr < RNODES; rr += 4)
            acc += prob[rr] * P[(size_t)rr * ODIM + o];   // coalesced 256B per r
        spart[g * ODIM + o] = acc;
        __syncthreads();
        if (g == 0) {
            float s = (spart[o] + spart[ODIM + o]) + (spart[2 * ODIM + o] + spart[3 * ODIM + o]);
            s *= inv;
            svec[o] = s;
            red[o]  = s * s;   // t == o here
        }
        __syncthreads();
        // reduce red[0..63] -> |s|^2
        for (int s = 32; s > 0; s >>= 1) {
            if (t < s) red[t] += red[t + s];
            __syncthreads();
        }
        const float sq = red[0];
        // squash: out = s * sq / ((1+sq) * sqrt(sq))   (no epsilon, per reference)
        if (t < ODIM) ovec[t] = svec[t] * (sq / ((1.f + sq) * sqrtf(sq)));
        __syncthreads();

        // ---- logits[r] += P[r][:] . out ----
        if (iter < 2) {
            for (int rr = t; rr < RNODES; rr += 256) {
                const float4* Pr = (const float4*)(P + (size_t)rr * ODIM);
                float d = 0.f;
                #pragma unroll
                for (int oo = 0; oo < ODIM / 4; oo++) {
                    const float4 pv = Pr[oo];
                    d += pv.x * ovec[oo * 4 + 0] + pv.y * ovec[oo * 4 + 1]
                       + pv.z * ovec[oo * 4 + 2] + pv.w * ovec[oo * 4 + 3];
                }
                logit[rr] += d;
            }
            __syncthreads();
        }
    }

    if (t < ODIM) out[(size_t)(c * BATCH + b) * ODIM + t] = ovec[t];
}

// ---------------------------------------------------------------------------
extern "C" void kernel_launch(void* const* d_in, const int* in_sizes, int n_in,
                              void* d_out, int out_size, void* d_ws, size_t ws_size,
                              hipStream_t stream) {
    const float* x  = (const float*)d_in[0];               // (64, 1152, 64)
    const float* w  = (const float*)d_in[1];               // (16, 1152, 64, 64)
    const float* rl = (const float*)d_in[2];               // (16, 1, 1152, 1, 1)
    float* out      = (float*)d_out;                       // (16, 64, 1, 1, 64)
    float* priors   = (float*)d_ws;                        // 16*64*1152*64 fp32 = 302 MB

    caps_priors_gemm<<<dim3(NCAPS * RNODES), dim3(128), 0, stream>>>(x, w, priors);
    caps_routing   <<<dim3(NCAPS * BATCH),  dim3(256), 0, stream>>>(priors, rl, out);
}